// HingeLoss_36679020708303
// MI455X (gfx1250) — compile-verified
//
#include <hip/hip_runtime.h>
#include <hip/hip_bf16.h>

typedef __attribute__((ext_vector_type(2))) float v2f;
typedef __attribute__((ext_vector_type(8))) float v8f;

#define BB 512
#define FF 256
#define EPSF 1e-8f
#define MSCALE 0.15f

// ---------------------------------------------------------------------------
// Kernel 1: per-row L2 norms with the torch eps scheme:
//   norm[i] = max(sqrt(sum(feats[i]^2) + eps), eps)
// One 256-thread block per row (F == 256).
// ---------------------------------------------------------------------------
__global__ void norms_kernel(const float* __restrict__ feats,
                             float* __restrict__ norm) {
    __shared__ float red[FF];
    const int row = blockIdx.x;
    const int t   = threadIdx.x;
    const float v = feats[row * FF + t];
    red[t] = v * v;
    __syncthreads();
    for (int s = FF / 2; s > 0; s >>= 1) {
        if (t < s) red[t] += red[t + s];
        __syncthreads();
    }
    if (t == 0) {
        norm[row] = fmaxf(sqrtf(red[0] + EPSF), EPSF);
    }
}

// ---------------------------------------------------------------------------
// Kernel 2: dot = feats @ feats^T via V_WMMA_F32_16X16X4_F32.
// One wave (32 threads) per 16x16 output tile; grid = (32, 32).
// A (16x4 f32, 2 VGPRs): lanes 0-15 hold M=lane, K = {k,k+1};
//                        lanes 16-31 hold M=lane-16, K = {k+2,k+3}.
// B (4x16 f32, 2 VGPRs): mirrored K split, N = lane&15. Since B = feats^T,
//                        B[k][n] = feats[n][k] -> same load formula as A with
//                        the column-tile row base.
// D (16x16 f32, 8 VGPRs): VGPR r -> rows r (lanes 0-15) and r+8 (lanes 16-31).
// blockDim = 32 => EXEC is all ones inside WMMA, as the ISA requires.
// ---------------------------------------------------------------------------
__global__ __launch_bounds__(32) void gemm16x16x4(const float* __restrict__ feats,
                                                  float* __restrict__ dot) {
    const int lane = threadIdx.x;
    const int half = lane >> 4;   // 0: K pair {0,1}; 1: K pair {2,3}
    const int l15  = lane & 15;

    const int rowA = blockIdx.x * 16 + l15;   // output row within tile
    const int rowB = blockIdx.y * 16 + l15;   // output col within tile
    const float* __restrict__ pa = feats + rowA * FF;
    const float* __restrict__ pb = feats + rowB * FF;

    v8f acc = {};
    for (int k = 0; k < FF; k += 4) {
        const int off = k + 2 * half;
        v2f a, b;
        a.x = pa[off];
        a.y = pa[off + 1];
        b.x = pb[off];
        b.y = pb[off + 1];
        // 8 args: (neg_a, A, neg_b, B, c_mod, C, reuse_a, reuse_b)
        acc = __builtin_amdgcn_wmma_f32_16x16x4_f32(
            false, a, false, b, (short)0, acc, false, false);
    }

    const int col = blockIdx.y * 16 + l15;
#pragma unroll
    for (int r = 0; r < 8; ++r) {
        const int row = blockIdx.x * 16 + r + 8 * half;
        dot[row * BB + col] = acc[r];
    }
}

// ---------------------------------------------------------------------------
// Kernel 3: per-row hinge reduction. One 256-thread block per row i.
//   cvals[j] = cos(i,j); simM/difM masks (off-diagonal); a[k] = marg(i,k)+cos(i,k)
//   rowSum  = sum_{k in dif} sum_{j in sim} relu(a[k] - cvals[j])
//   rowLoss = valid ? rowSum / (|sim|*|dif|) : 0 ; rowValid = valid ? 1 : 0
// ---------------------------------------------------------------------------
__global__ void rowloss_kernel(const float* __restrict__ dot,
                               const float* __restrict__ norm,
                               const int* __restrict__ ids,
                               float* __restrict__ rowLoss,
                               float* __restrict__ rowValid) {
    __shared__ float cvals[BB];
    __shared__ float avals[BB];
    __shared__ float simM[BB];
    __shared__ float difM[BB];
    __shared__ float redS[256];
    __shared__ float redC1[256];
    __shared__ float redC2[256];

    const int i = blockIdx.x;
    const int t = threadIdx.x;

    const int id0 = ids[i * 4 + 0];
    const int id1 = ids[i * 4 + 1];
    const int id2 = ids[i * 4 + 2];
    const int id3 = ids[i * 4 + 3];
    const float ni = norm[i];

    for (int j = t; j < BB; j += 256) {
        const float c = dot[i * BB + j] / (ni * norm[j]);
        const int l1 = abs(id0 - ids[j * 4 + 0]) + abs(id1 - ids[j * 4 + 1]) +
                       abs(id2 - ids[j * 4 + 2]) + abs(id3 - ids[j * 4 + 3]);
        const bool eq   = (l1 == 0);               // integer ids: L1==0 <=> all equal
        const bool diag = (j == i);
        cvals[j] = c;
        avals[j] = MSCALE * (float)l1 + c;         // marg(i,k) + cos(i,k)
        simM[j]  = (eq && !diag) ? 1.0f : 0.0f;
        difM[j]  = (!eq && !diag) ? 1.0f : 0.0f;
    }
    __syncthreads();

    float sum = 0.0f, scnt = 0.0f, dcnt = 0.0f;
    for (int k = t; k < BB; k += 256) {
        scnt += simM[k];
        dcnt += difM[k];
        if (difM[k] != 0.0f) {
            const float a = avals[k];
            float s = 0.0f;
#pragma unroll 4
            for (int j = 0; j < BB; ++j) {
                s += simM[j] * fmaxf(a - cvals[j], 0.0f);
            }
            sum += s;
        }
    }
    redS[t]  = sum;
    redC1[t] = scnt;
    redC2[t] = dcnt;
    __syncthreads();
    for (int s = 128; s > 0; s >>= 1) {
        if (t < s) {
            redS[t]  += redS[t + s];
            redC1[t] += redC1[t + s];
            redC2[t] += redC2[t + s];
        }
        __syncthreads();
    }
    if (t == 0) {
        const float cnt   = redC1[0] * redC2[0];
        const bool  valid = cnt > 0.0f;
        rowLoss[i]  = valid ? (redS[0] / cnt) : 0.0f;
        rowValid[i] = valid ? 1.0f : 0.0f;
    }
}

// ---------------------------------------------------------------------------
// Kernel 4: deterministic final reduction over 512 rows.
//   out = sum(rowLoss) / max(sum(rowValid), 1)
// ---------------------------------------------------------------------------
__global__ void finalize_kernel(const float* __restrict__ rowLoss,
                                const float* __restrict__ rowValid,
                                float* __restrict__ out) {
    __shared__ float s1[BB];
    __shared__ float s2[BB];
    const int t = threadIdx.x;
    s1[t] = rowLoss[t];
    s2[t] = rowValid[t];
    __syncthreads();
    for (int s = BB / 2; s > 0; s >>= 1) {
        if (t < s) {
            s1[t] += s1[t + s];
            s2[t] += s2[t + s];
        }
        __syncthreads();
    }
    if (t == 0) out[0] = s1[0] / fmaxf(s2[0], 1.0f);
}

// ---------------------------------------------------------------------------
// Host-side launch. Workspace layout (floats):
//   [0, BB*BB)                : dot = feats @ feats^T
//   [BB*BB, BB*BB+BB)         : norms
//   [+BB, +2*BB)              : rowLoss
//   [+2*BB, +3*BB)            : rowValid
// All workspace regions are fully written before being read every call.
// ---------------------------------------------------------------------------
extern "C" void kernel_launch(void* const* d_in, const int* in_sizes, int n_in,
                              void* d_out, int out_size, void* d_ws, size_t ws_size,
                              hipStream_t stream) {
    const int*   ids   = (const int*)d_in[0];     // [512, 4]
    const float* feats = (const float*)d_in[1];   // [512, 256]
    float*       out   = (float*)d_out;           // scalar

    float* ws       = (float*)d_ws;
    float* dot      = ws;                 // BB*BB
    float* norm     = ws + BB * BB;       // BB
    float* rowLoss  = norm + BB;          // BB
    float* rowValid = rowLoss + BB;       // BB

    norms_kernel<<<BB, FF, 0, stream>>>(feats, norm);
    gemm16x16x4<<<dim3(BB / 16, BB / 16), 32, 0, stream>>>(feats, dot);
    rowloss_kernel<<<BB, 256, 0, stream>>>(dot, norm, ids, rowLoss, rowValid);
    finalize_kernel<<<1, BB, 0, stream>>>(rowLoss, rowValid, out);
}